// TopQPooling_41120016892314
// MI455X (gfx1250) — compile-verified
//
#include <hip/hip_runtime.h>
#include <hip/hip_bf16.h>
#include <math.h>

// Problem constants (match reference)
#define BB 32
#define TT 2048
#define DD 512
#define KMAX 308          // ceil(2048 * 0.15) with f32 math = 308
#define KPAD 320
#define NEGV -1e9f

typedef __attribute__((ext_vector_type(2))) float v2f;
typedef __attribute__((ext_vector_type(8))) float v8f;

// ---------------------------------------------------------------------------
// Pass 1: scores[b,t] = (t < len[b]) ? ||H[b,t,:]||_2 : -1e9
// One wave32 per row; float4 (b128) coalesced loads; shfl_xor reduction.
// ---------------------------------------------------------------------------
__global__ __launch_bounds__(256) void topq_score_kernel(
    const float* __restrict__ H, const int* __restrict__ lengths,
    float* __restrict__ scores)
{
    const int wave = threadIdx.x >> 5;
    const int lane = threadIdx.x & 31;
    const int row  = blockIdx.x * 8 + wave;     // row in [0, B*T)
    const int b    = row >> 11;                 // / 2048
    const int t    = row & (TT - 1);

    const float* p = H + (size_t)row * DD;
    float s = 0.0f;
#pragma unroll
    for (int it = 0; it < 4; ++it) {
        float4 v = *(const float4*)(p + it * 128 + lane * 4);
        s += v.x * v.x + v.y * v.y + v.z * v.z + v.w * v.w;
    }
#pragma unroll
    for (int m = 16; m >= 1; m >>= 1)
        s += __shfl_xor(s, m, 32);

    if (lane == 0) {
        const int len = lengths[b];
        scores[row] = (t < len) ? sqrtf(s) : NEGV;
    }
}

// ---------------------------------------------------------------------------
// Pass 2: per-b stable top-k selection via rank counting (no atomics).
// rank(t) = #{s : sc[s] > sc[t]} + #{s < t : sc[s] == sc[t]}  -> unique ranks,
// identical ordering to lax.top_k (descending, stable by index).
// Selected timesteps written in rank order: sel[b*KPAD + rank] = t.
// ---------------------------------------------------------------------------
__global__ __launch_bounds__(1024) void topq_select_kernel(
    const float* __restrict__ scores, const int* __restrict__ lengths,
    int* __restrict__ sel)
{
    __shared__ float sc[TT];
    const int b   = blockIdx.x;
    const int tid = threadIdx.x;

    for (int t = tid; t < TT; t += 1024)
        sc[t] = scores[b * TT + t];
    __syncthreads();

    const int len = lengths[b];
    int k = (int)ceilf((float)len * 0.15f);   // exact f32 semantics of reference
    if (k < 1) k = 1;

    for (int t = tid; t < TT; t += 1024) {
        const float my = sc[t];
        int rank = 0;
        for (int s = 0; s < TT; ++s) {
            const float v = sc[s];
            rank += (v > my) || ((v == my) && (s < t));
        }
        if (rank < k)
            sel[b * KPAD + rank] = t;
    }
}

// ---------------------------------------------------------------------------
// Pass 3: pooled[b,:] = (1/k) * sum over selected rows, via WMMA f32 16x16x4.
// A = all-ones (16x4)  =>  D[m,n] = sum_K B[K,n]  (column sum), independent of
// the hardware's K<->slot assignment in the B operand.  B operand slots
// (vgpr e, lane-half h) carry row sel[i + 2e + h], columns = lane&15.
// sidx[] is zero-padded past k so the inner loop is fully branchless:
// unconditional ds_load of a valid row index, contribution zeroed by mask.
// One block per b; 16 waves x 2 column tiles = 512 cols.
// ---------------------------------------------------------------------------
__global__ __launch_bounds__(512) void topq_pool_kernel(
    const float* __restrict__ H, const int* __restrict__ lengths,
    const int* __restrict__ sel, float* __restrict__ out)
{
    __shared__ int sidx[KPAD];
    const int b   = blockIdx.x;
    const int tid = threadIdx.x;

    const int len = lengths[b];
    int k = (int)ceilf((float)len * 0.15f);
    if (k < 1) k = 1;

    // Zero-pad beyond k: every sidx[] entry is a VALID row index (0 if padded),
    // so the hot loop never needs a select on the index nor a branch.
    if (tid < KPAD)
        sidx[tid] = (tid < k) ? sel[b * KPAD + tid] : 0;
    __syncthreads();

    const int lane = tid & 31;
    const int half = lane >> 4;     // 0 or 1
    const int col  = lane & 15;
    const int wave = tid >> 5;      // 0..15
    const int c0   = wave * 32;     // this wave owns columns [c0, c0+32)

    const float* Hb = H + (size_t)b * TT * DD;

    v2f a; a[0] = 1.0f; a[1] = 1.0f;   // all-ones A matrix
    v8f acc0 = {};
    v8f acc1 = {};

    for (int i = 0; i < k; i += 4) {
        const int j0 = i + half;        // row slot for B vgpr 0
        const int j1 = i + 2 + half;    // row slot for B vgpr 1
        const int r0 = sidx[j0];        // unconditional, always in [0,T)
        const int r1 = sidx[j1];
        const float m0 = (j0 < k) ? 1.0f : 0.0f;
        const float m1 = (j1 < k) ? 1.0f : 0.0f;

        const float* q0 = Hb + (size_t)r0 * DD + c0 + col;
        const float* q1 = Hb + (size_t)r1 * DD + c0 + col;

        v2f b0, b1;
        b0[0] = q0[0]  * m0;  b0[1] = q1[0]  * m1;   // tile [c0, c0+16)
        b1[0] = q0[16] * m0;  b1[1] = q1[16] * m1;   // tile [c0+16, c0+32)

        acc0 = __builtin_amdgcn_wmma_f32_16x16x4_f32(
            false, a, false, b0, (short)0, acc0, false, false);
        acc1 = __builtin_amdgcn_wmma_f32_16x16x4_f32(
            false, a, false, b1, (short)0, acc1, false, false);
    }

    const float inv = 1.0f / (float)k;
    if (half == 0) {
        // Every M row of D is identical (A all-ones); lane n holds column n.
        out[b * DD + c0 + col]      = acc0[0] * inv;
        out[b * DD + c0 + 16 + col] = acc1[0] * inv;
    }
}

// ---------------------------------------------------------------------------
extern "C" void kernel_launch(void* const* d_in, const int* in_sizes, int n_in,
                              void* d_out, int out_size, void* d_ws, size_t ws_size,
                              hipStream_t stream)
{
    (void)in_sizes; (void)n_in; (void)out_size; (void)ws_size;

    const float* H       = (const float*)d_in[0];
    const int*   lengths = (const int*)d_in[1];
    float*       out     = (float*)d_out;

    float* scores = (float*)d_ws;                                  // B*T floats (256 KB)
    int*   sel    = (int*)((char*)d_ws + (size_t)BB * TT * 4);     // B*KPAD ints (40 KB)

    topq_score_kernel<<<dim3(BB * TT / 8), dim3(256), 0, stream>>>(H, lengths, scores);
    topq_select_kernel<<<dim3(BB), dim3(1024), 0, stream>>>(scores, lengths, sel);
    topq_pool_kernel<<<dim3(BB), dim3(512), 0, stream>>>(H, lengths, sel, out);
}